// SpatialMix_HPCM_74397423501646
// MI455X (gfx1250) — compile-verified
//
#include <hip/hip_runtime.h>
#include <cstddef>

#define BB 8
#define CC 192
#define HH 64
#define WW 64
#define TT (HH * WW)            // 4096
#define BT (BB * TT)            // 32768 rows of flattened activation matrix
#define CL 64                   // scan chunk length
#define NC (TT / CL)            // 64 chunks per (b,c) sequence

typedef __attribute__((ext_vector_type(2))) float v2f;
typedef __attribute__((ext_vector_type(8))) float v8f;

// ---------------------------------------------------------------------------
// K1: omni_shift fused with NCHW -> (B*T, C) transpose.
// ---------------------------------------------------------------------------
__global__ __launch_bounds__(256) void omni_shift_k(
    const float* __restrict__ x, const float* __restrict__ alpha,
    const float* __restrict__ dw1, const float* __restrict__ dw3,
    const float* __restrict__ dw5, float* __restrict__ xs)
{
    int idx = blockIdx.x * blockDim.x + threadIdx.x;
    const int total = BB * CC * HH * WW;
    if (idx >= total) return;

    int w = idx % WW;
    int h = (idx / WW) % HH;
    int c = (idx / (WW * HH)) % CC;
    int b = idx / (WW * HH * CC);

    const float* xp = x + ((size_t)(b * CC + c)) * (HH * WW);
    float center = xp[h * WW + w];

    float s3 = 0.0f;
    #pragma unroll
    for (int i = 0; i < 3; ++i) {
        int hh = h + i - 1;
        if ((unsigned)hh >= (unsigned)HH) continue;
        #pragma unroll
        for (int j = 0; j < 3; ++j) {
            int ww = w + j - 1;
            if ((unsigned)ww >= (unsigned)WW) continue;
            s3 += dw3[c * 9 + i * 3 + j] * xp[hh * WW + ww];
        }
    }

    float s5 = 0.0f;
    #pragma unroll
    for (int i = 0; i < 5; ++i) {
        int hh = h + i - 2;
        if ((unsigned)hh >= (unsigned)HH) continue;
        #pragma unroll
        for (int j = 0; j < 5; ++j) {
            int ww = w + j - 2;
            if ((unsigned)ww >= (unsigned)WW) continue;
            s5 += dw5[c * 25 + i * 5 + j] * xp[hh * WW + ww];
        }
    }

    float a0 = alpha[0], a1 = alpha[1], a2 = alpha[2], a3 = alpha[3];
    float val = a0 * center + a1 * dw1[c] * center + a2 * s3 + a3 * s5;
    xs[((size_t)b * TT + h * WW + w) * CC + c] = val;
}

// ---------------------------------------------------------------------------
// K2: k/v/sr projections via V_WMMA_F32_16X16X4_F32 (one wave per 16x16 tile).
// ---------------------------------------------------------------------------
__global__ __launch_bounds__(32) void proj_gemm_k(
    const float* __restrict__ xs,
    const float* __restrict__ wk, const float* __restrict__ wv,
    const float* __restrict__ wr,
    float* __restrict__ ok, float* __restrict__ ov, float* __restrict__ osr)
{
    const int l    = threadIdx.x;
    const int half = l >> 4;
    const int lm   = l & 15;
    const int m0   = blockIdx.x * 16;
    const int d0   = blockIdx.y * 16;
    const int z    = blockIdx.z;

    const float* w   = (z == 0) ? wk : (z == 1) ? wv : wr;
    float*       out = (z == 0) ? ok : (z == 1) ? ov : osr;

    const float* arow = xs + (size_t)(m0 + lm) * CC + half * 2;
    const float* brow = w  + (size_t)(d0 + lm) * CC + half * 2;

    v8f acc = {0.f, 0.f, 0.f, 0.f, 0.f, 0.f, 0.f, 0.f};
    #pragma unroll
    for (int kk = 0; kk < CC / 4; ++kk) {
        v2f a = *(const v2f*)(arow + kk * 4);
        v2f b = *(const v2f*)(brow + kk * 4);
        acc = __builtin_amdgcn_wmma_f32_16x16x4_f32(
            false, a, false, b, (short)0, acc, false, false);
    }

    const bool sig = (z == 2);
    #pragma unroll
    for (int r = 0; r < 8; ++r) {
        float vo = acc[r];
        if (sig) vo = 1.0f / (1.0f + __expf(-vo));
        out[(size_t)(m0 + r + half * 8) * CC + d0 + lm] = vo;
    }
}

// ---------------------------------------------------------------------------
// WKV scan, chunked (associative in log-sum-exp semiring).
// One-element step:  q = max(p-w, k); n' = e^{p-w-q} n + e^{k-q} v; p' = q.
// Chunk of length L acts as: state' = combine(decay(state, L*w), chunk_agg).
// ---------------------------------------------------------------------------

// Phase 1: per-chunk aggregates (blockIdx.y: 0 = forward, 1 = backward).
__global__ __launch_bounds__(256) void wkv_chunk_agg_k(
    const float* __restrict__ kb, const float* __restrict__ vb,
    const float* __restrict__ decay,
    float* __restrict__ aggP, float* __restrict__ aggN, float* __restrict__ aggD)
{
    int tid = blockIdx.x * blockDim.x + threadIdx.x;
    const int total = BB * CC * NC;
    if (tid >= total) return;
    const int dir = blockIdx.y;
    int c = tid % CC;
    int j = (tid / CC) % NC;
    int b = tid / (CC * NC);
    float wv = decay[c] / (float)TT;

    float p = -INFINITY, n = 0.0f, d = 0.0f;
    for (int i = 0; i < CL; ++i) {
        int t = j * CL + (dir == 0 ? i : (CL - 1 - i));
        size_t base = ((size_t)b * TT + t) * CC + c;
        float kt = kb[base];
        float vt = vb[base];
        float q  = fmaxf(p - wv, kt);
        float e1 = __expf(p - wv - q);
        float e2 = __expf(kt - q);
        n = e1 * n + e2 * vt;
        d = e1 * d + e2;
        p = q;
    }
    size_t ai = (((size_t)dir * BB + b) * NC + j) * CC + c;
    aggP[ai] = p; aggN[ai] = n; aggD[ai] = d;
}

// Phase 2: exclusive prefix over chunk aggregates (per (b,c), per direction).
__global__ __launch_bounds__(256) void wkv_chunk_prefix_k(
    const float* __restrict__ aggP, const float* __restrict__ aggN,
    const float* __restrict__ aggD, const float* __restrict__ decay,
    float* __restrict__ Sp, float* __restrict__ Sn, float* __restrict__ Sd)
{
    int tid = blockIdx.x * blockDim.x + threadIdx.x;
    if (tid >= BB * CC) return;
    const int dir = blockIdx.y;
    int b = tid / CC;
    int c = tid % CC;
    float Lw = (decay[c] / (float)TT) * (float)CL;

    float p = -INFINITY, n = 0.0f, d = 0.0f;
    for (int s = 0; s < NC; ++s) {
        int j = (dir == 0) ? s : (NC - 1 - s);
        size_t ai = (((size_t)dir * BB + b) * NC + j) * CC + c;
        Sp[ai] = p; Sn[ai] = n; Sd[ai] = d;   // exclusive state before chunk j
        float p2 = aggP[ai], n2 = aggN[ai], d2 = aggD[ai];
        float p1 = p - Lw;
        float q  = fmaxf(p1, p2);
        float e1 = __expf(p1 - q);
        float e2 = __expf(p2 - q);
        n = e1 * n + e2 * n2;
        d = e1 * d + e2 * d2;
        p = q;
    }
}

// Phase 3a: forward emit — per-element exclusive forward states.
__global__ __launch_bounds__(256) void wkv_fwd_emit_k(
    const float* __restrict__ kb, const float* __restrict__ vb,
    const float* __restrict__ decay,
    const float* __restrict__ Sp, const float* __restrict__ Sn,
    const float* __restrict__ Sd,
    float* __restrict__ pf, float* __restrict__ nf, float* __restrict__ df)
{
    int tid = blockIdx.x * blockDim.x + threadIdx.x;
    const int total = BB * CC * NC;
    if (tid >= total) return;
    int c = tid % CC;
    int j = (tid / CC) % NC;
    int b = tid / (CC * NC);
    float wv = decay[c] / (float)TT;

    size_t si = (((size_t)0 * BB + b) * NC + j) * CC + c;
    float p = Sp[si], n = Sn[si], d = Sd[si];
    for (int i = 0; i < CL; ++i) {
        int t = j * CL + i;
        size_t base = ((size_t)b * TT + t) * CC + c;
        pf[base] = p; nf[base] = n; df[base] = d;
        float kt = kb[base];
        float vt = vb[base];
        float q  = fmaxf(p - wv, kt);
        float e1 = __expf(p - wv - q);
        float e2 = __expf(kt - q);
        n = e1 * n + e2 * vt;
        d = e1 * d + e2;
        p = q;
    }
}

// Phase 3b: backward emit fused with the 3-way combine + receptance gate.
__global__ __launch_bounds__(256) void wkv_bwd_combine_k(
    const float* __restrict__ kb, const float* __restrict__ vb,
    const float* __restrict__ sr,
    const float* __restrict__ pf, const float* __restrict__ nf,
    const float* __restrict__ df,
    const float* __restrict__ Sp, const float* __restrict__ Sn,
    const float* __restrict__ Sd,
    const float* __restrict__ decay, const float* __restrict__ boost,
    float* __restrict__ yb)
{
    int tid = blockIdx.x * blockDim.x + threadIdx.x;
    const int total = BB * CC * NC;
    if (tid >= total) return;
    int c = tid % CC;
    int j = (tid / CC) % NC;
    int b = tid / (CC * NC);
    float wv = decay[c] / (float)TT;
    float u  = boost[c] / (float)TT;

    size_t si = (((size_t)1 * BB + b) * NC + j) * CC + c;
    float p = Sp[si], n = Sn[si], d = Sd[si];
    for (int i = 0; i < CL; ++i) {
        int t = j * CL + (CL - 1 - i);
        size_t base = ((size_t)b * TT + t) * CC + c;
        float kt  = kb[base];
        float vt  = vb[base];
        float pfv = pf[base], nfv = nf[base], dfv = df[base];

        float ps = u + kt;
        float r  = fmaxf(fmaxf(pfv, p), ps);   // ps finite => r finite
        float ef = __expf(pfv - r);
        float eb = __expf(p - r);
        float es = __expf(ps - r);
        float y  = (ef * nfv + eb * n + es * vt) / (ef * dfv + eb * d + es);
        yb[base] = sr[base] * y;

        float q  = fmaxf(p - wv, kt);
        float e1 = __expf(p - wv - q);
        float e2 = __expf(kt - q);
        n = e1 * n + e2 * vt;
        d = e1 * d + e2;
        p = q;
    }
}

// ---------------------------------------------------------------------------
// K5: output projection via WMMA, writing transposed straight into NCHW.
// ---------------------------------------------------------------------------
__global__ __launch_bounds__(32) void out_gemm_k(
    const float* __restrict__ yb, const float* __restrict__ wo,
    float* __restrict__ out)
{
    const int l    = threadIdx.x;
    const int half = l >> 4;
    const int lm   = l & 15;
    const int m0   = blockIdx.x * 16;
    const int d0   = blockIdx.y * 16;

    const float* arow = yb + (size_t)(m0 + lm) * CC + half * 2;
    const float* brow = wo + (size_t)(d0 + lm) * CC + half * 2;

    v8f acc = {0.f, 0.f, 0.f, 0.f, 0.f, 0.f, 0.f, 0.f};
    #pragma unroll
    for (int kk = 0; kk < CC / 4; ++kk) {
        v2f a = *(const v2f*)(arow + kk * 4);
        v2f b = *(const v2f*)(brow + kk * 4);
        acc = __builtin_amdgcn_wmma_f32_16x16x4_f32(
            false, a, false, b, (short)0, acc, false, false);
    }

    const int bidx = m0 / TT;
    const int t0   = m0 - bidx * TT;
    #pragma unroll
    for (int r = 0; r < 8; ++r) {
        int t = t0 + r + half * 8;
        out[(size_t)bidx * CC * TT + (size_t)(d0 + lm) * TT + t] = acc[r];
    }
}

// ---------------------------------------------------------------------------
extern "C" void kernel_launch(void* const* d_in, const int* in_sizes, int n_in,
                              void* d_out, int out_size, void* d_ws, size_t ws_size,
                              hipStream_t stream) {
    (void)in_sizes; (void)n_in; (void)out_size; (void)ws_size;

    const float* x        = (const float*)d_in[0];
    const float* w_key    = (const float*)d_in[1];
    const float* w_value  = (const float*)d_in[2];
    const float* w_recept = (const float*)d_in[3];
    const float* w_out    = (const float*)d_in[4];
    const float* decay    = (const float*)d_in[5];
    const float* boost    = (const float*)d_in[6];
    const float* alpha    = (const float*)d_in[7];
    const float* dw1      = (const float*)d_in[8];
    const float* dw3      = (const float*)d_in[9];
    const float* dw5      = (const float*)d_in[10];

    float* out = (float*)d_out;
    float* ws  = (float*)d_ws;

    const size_t N = (size_t)BT * CC;           // one (B*T, C) array
    const size_t M = (size_t)2 * BB * NC * CC;  // chunk stats, both directions

    float* xs   = ws;          // reused as y after projections
    float* kb   = ws + 1 * N;
    float* vb   = ws + 2 * N;
    float* sb   = ws + 3 * N;
    float* pf   = ws + 4 * N;
    float* nf   = ws + 5 * N;
    float* df   = ws + 6 * N;
    float* aggP = ws + 7 * N;
    float* aggN = aggP + M;
    float* aggD = aggN + M;
    float* Sp   = aggD + M;
    float* Sn   = Sp + M;
    float* Sd   = Sn + M;
    float* yb   = xs;

    // K1: omni_shift + transpose
    {
        int total = BB * CC * HH * WW;
        omni_shift_k<<<(total + 255) / 256, 256, 0, stream>>>(
            x, alpha, dw1, dw3, dw5, xs);
    }
    // K2: k, v, sr projections (WMMA)
    proj_gemm_k<<<dim3(BT / 16, CC / 16, 3), 32, 0, stream>>>(
        xs, w_key, w_value, w_recept, kb, vb, sb);

    // K3: chunked bidirectional WKV scan
    {
        int totalChunks = BB * CC * NC;
        wkv_chunk_agg_k<<<dim3((totalChunks + 255) / 256, 2), 256, 0, stream>>>(
            kb, vb, decay, aggP, aggN, aggD);
        wkv_chunk_prefix_k<<<dim3((BB * CC + 255) / 256, 2), 256, 0, stream>>>(
            aggP, aggN, aggD, decay, Sp, Sn, Sd);
        wkv_fwd_emit_k<<<(totalChunks + 255) / 256, 256, 0, stream>>>(
            kb, vb, decay, Sp, Sn, Sd, pf, nf, df);
        wkv_bwd_combine_k<<<(totalChunks + 255) / 256, 256, 0, stream>>>(
            kb, vb, sb, pf, nf, df, Sp, Sn, Sd, decay, boost, yb);
    }

    // K5: output projection (WMMA) + transpose to NCHW
    out_gemm_k<<<dim3(BT / 16, CC / 16, 1), 32, 0, stream>>>(yb, w_out, out);
}